// InterpretableMultiHeadAttention_10101763080304
// MI455X (gfx1250) — compile-verified
//
#include <hip/hip_runtime.h>
#include <hip/hip_bf16.h>

// ---------------------------------------------------------------------------
// InterpretableMultiHeadAttention on gfx1250 (CDNA5, wave32, WMMA f32).
// All GEMMs run on the matrix pipe via V_WMMA_F32_16X16X4_F32 (full fp32 on
// the softmax path; avg_attn is a validated output).  All LDS operand tiles
// are laid out so each WMMA fragment is ONE contiguous, 8B-aligned pair ->
// single ds_load_b64 into an even VGPR pair (no v_mov repacking).
// ---------------------------------------------------------------------------

#define BATCH 8
#define SEQ   2048
#define DMODEL 256
#define NHEAD 4
#define DK    64
#define INV_SCALE 0.125f   // 1/sqrt(64)

typedef float v2f __attribute__((ext_vector_type(2)));
typedef float v8f __attribute__((ext_vector_type(8)));

__device__ inline v8f v8f_zero() {
  v8f z = {0.f, 0.f, 0.f, 0.f, 0.f, 0.f, 0.f, 0.f};
  return z;
}

// D = A(16x4) * B(4x16) + C(16x16), fp32 everywhere.
__device__ inline v8f wmma_f32(v2f a, v2f b, v8f c) {
  return __builtin_amdgcn_wmma_f32_16x16x4_f32(
      /*neg_a=*/false, a, /*neg_b=*/false, b,
      /*c_mod=*/(short)0, c, /*reuse_a=*/false, /*reuse_b=*/false);
}

// Fragment pair load from an LDS tile stored [row][k] with k contiguous.
//   A-matrix 16x4:  row = M (lane 0-15), lane-half selects K base +2.
//   B-matrix 4x16 from a TRANSPOSED tile [n][k]: row = N, same K selection.
// `stride` must be even so the pair is 8-byte aligned (ds_load_b64).
__device__ inline v2f ldPair(const float* base, int stride, int row, int k, int hl) {
  v2f a;
  a.x = base[row * stride + k + 2 * hl + 0];
  a.y = base[row * stride + k + 2 * hl + 1];
  return a;
}

__device__ inline void softmerge(float& m, float& s, float m2, float s2) {
  float mn = fmaxf(m, m2);
  s = s * __expf(m - mn) + s2 * __expf(m2 - mn);
  m = mn;
}

// ---------------------------------------------------------------------------
// C[M x 256] = X[M x 256] @ W[256 x 256] + bias.  grid.x = M/64, block = 256.
// Block tile 64x256; wave w: rows (w&3)*16, cols (w>>2)*128 (8 n-tiles).
// W tile staged TRANSPOSED in LDS: Wt[n][k].
// ---------------------------------------------------------------------------
__global__ __launch_bounds__(256) void gemm_xw(const float* __restrict__ X,
                                               const float* __restrict__ W,
                                               const float* __restrict__ bias,
                                               float* __restrict__ C) {
  __shared__ float Xs[64][18];    // [m][k], even pad: conflict-free b64 frags
  __shared__ float Wt[256][18];   // [n][k] transposed

  const int tid = threadIdx.x;
  const int lane = tid & 31;
  const int wv = tid >> 5;
  const int mr = lane & 15;
  const int hl = lane >> 4;
  const int row0 = blockIdx.x * 64;
  const int m0 = (wv & 3) * 16;
  const int nb = (wv >> 2) * 128;

  v8f acc[8];
#pragma unroll
  for (int j = 0; j < 8; ++j) acc[j] = v8f_zero();

  for (int k0 = 0; k0 < DMODEL; k0 += 16) {
    {  // X tile 64x16, row-major
      int r = tid >> 2, c4 = (tid & 3) * 4;
      const float4 x = *(const float4*)(X + (size_t)(row0 + r) * DMODEL + k0 + c4);
      Xs[r][c4 + 0] = x.x; Xs[r][c4 + 1] = x.y;
      Xs[r][c4 + 2] = x.z; Xs[r][c4 + 3] = x.w;
      if (k0 + 16 < DMODEL)  // hide next-tile latency behind this tile's WMMAs
        __builtin_prefetch(X + (size_t)(row0 + r) * DMODEL + k0 + 16 + c4, 0, 3);
    }
#pragma unroll
    for (int j = 0; j < 4; ++j) {  // W tile 16x256 -> transposed store
      int idx = tid + 256 * j;
      int r = idx >> 6, c4 = (idx & 63) * 4;
      const float4 w4 = *(const float4*)(W + (size_t)(k0 + r) * DMODEL + c4);
      Wt[c4 + 0][r] = w4.x; Wt[c4 + 1][r] = w4.y;
      Wt[c4 + 2][r] = w4.z; Wt[c4 + 3][r] = w4.w;
      if (k0 + 16 < DMODEL)
        __builtin_prefetch(W + (size_t)(k0 + 16 + r) * DMODEL + c4, 0, 3);
    }
    __syncthreads();
#pragma unroll
    for (int ks = 0; ks < 16; ks += 4) {
      v2f a = ldPair(&Xs[0][0], 18, m0 + mr, ks, hl);
#pragma unroll
      for (int j = 0; j < 8; ++j) {
        v2f b = ldPair(&Wt[0][0], 18, nb + j * 16 + mr, ks, hl);
        acc[j] = wmma_f32(a, b, acc[j]);
      }
    }
    __syncthreads();
  }

#pragma unroll
  for (int j = 0; j < 8; ++j) {
    int n = nb + j * 16 + mr;
    float bv = bias[n];
#pragma unroll
    for (int v = 0; v < 8; ++v) {
      C[(size_t)(row0 + m0 + v + 8 * hl) * DMODEL + n] = acc[j][v] + bv;
    }
  }
}

// ---------------------------------------------------------------------------
// Softmax stats: per (b, h, 16 q-rows) compute row max m and exp-sum Z over
// all 2048 keys.  8 waves each cover 256 keys with online (m,s) accumulation.
// grid = (SEQ/16, NHEAD, BATCH), block = 256.
// ---------------------------------------------------------------------------
__global__ __launch_bounds__(256) void attn_stats(const float* __restrict__ Qp,
                                                  const float* __restrict__ Kp,
                                                  float* __restrict__ wm,
                                                  float* __restrict__ wz) {
  __shared__ float Qs[16][68];      // [q][k] even stride: b64 frags
  __shared__ float Ks[8][16][68];   // [key][k] per wave (K^T frags = row-major pairs)
  __shared__ float redm[8][32][8];
  __shared__ float reds[8][32][8];

  const int tid = threadIdx.x;
  const int lane = tid & 31;
  const int wv = tid >> 5;
  const int mr = lane & 15;
  const int hl = lane >> 4;
  const int q0 = blockIdx.x * 16;
  const int h = blockIdx.y;
  const int b = blockIdx.z;

  {  // Q head-slice 16x64
    int r = tid >> 4, c4 = (tid & 15) * 4;
    const float4 q = *(const float4*)(Qp + (size_t)(b * SEQ + q0 + r) * DMODEL + h * DK + c4);
    Qs[r][c4 + 0] = q.x; Qs[r][c4 + 1] = q.y;
    Qs[r][c4 + 2] = q.z; Qs[r][c4 + 3] = q.w;
  }
  __syncthreads();

  float mo[8], so[8];
#pragma unroll
  for (int v = 0; v < 8; ++v) { mo[v] = -1e30f; so[v] = 0.f; }

  for (int kt = 0; kt < 16; ++kt) {
    const int key0 = wv * 256 + kt * 16;
#pragma unroll
    for (int j = 0; j < 8; ++j) {  // wave-private K tile 16x64
      int idx = lane + 32 * j;
      int r = idx >> 4, c4 = (idx & 15) * 4;
      const float4 k4 = *(const float4*)(Kp + (size_t)(b * SEQ + key0 + r) * DMODEL + h * DK + c4);
      Ks[wv][r][c4 + 0] = k4.x; Ks[wv][r][c4 + 1] = k4.y;
      Ks[wv][r][c4 + 2] = k4.z; Ks[wv][r][c4 + 3] = k4.w;
    }
    __syncthreads();

    v8f s = v8f_zero();
#pragma unroll
    for (int ks = 0; ks < DK; ks += 4) {
      v2f a = ldPair(&Qs[0][0], 68, mr, ks, hl);
      v2f bt = ldPair(&Ks[wv][0][0], 68, mr, ks, hl);  // B = K^T
      s = wmma_f32(a, bt, s);
    }
    __syncthreads();

#pragma unroll
    for (int v = 0; v < 8; ++v) {  // online softmax update, key = key0 + mr
      float sc = s[v] * INV_SCALE;
      float mn = fmaxf(mo[v], sc);
      so[v] = so[v] * __expf(mo[v] - mn) + __expf(sc - mn);
      mo[v] = mn;
    }
  }

#pragma unroll
  for (int v = 0; v < 8; ++v) { redm[wv][lane][v] = mo[v]; reds[wv][lane][v] = so[v]; }
  __syncthreads();

  if (tid < 16) {  // reduce 8 waves x 16 key-columns per q-row
    int r = tid, lh = r >> 3, vv = r & 7;
    float m = -1e30f, sacc = 0.f;
    for (int w = 0; w < 8; ++w)
      for (int c = 0; c < 16; ++c)
        softmerge(m, sacc, redm[w][lh * 16 + c][vv], reds[w][lh * 16 + c][vv]);
    size_t o = (size_t)(b * NHEAD + h) * SEQ + q0 + r;
    wm[o] = m;
    wz[o] = sacc;
  }
}

// ---------------------------------------------------------------------------
// avg_attn tile: per (key-chunk 128, 16 q-rows, b) recompute scores per head,
// apply exp(s - m)/Z, average over heads, store to d_out avg region.
// grid = (SEQ/128, SEQ/16, BATCH), block = 256; wave w owns 16 keys.
// ---------------------------------------------------------------------------
__global__ __launch_bounds__(256) void attn_avg(const float* __restrict__ Qp,
                                                const float* __restrict__ Kp,
                                                const float* __restrict__ wm,
                                                const float* __restrict__ wz,
                                                float* __restrict__ avg) {
  __shared__ float Qs[16][260];     // all 4 heads of Q rows, even stride
  __shared__ float Ks[8][16][68];   // per-wave K slice (one head at a time)
  __shared__ float lm[4][16];
  __shared__ float lz[4][16];

  const int tid = threadIdx.x;
  const int lane = tid & 31;
  const int wv = tid >> 5;
  const int mr = lane & 15;
  const int hl = lane >> 4;
  const int kc0 = blockIdx.x * 128;
  const int q0 = blockIdx.y * 16;
  const int b = blockIdx.z;
  const int key0 = kc0 + wv * 16;

#pragma unroll
  for (int j = 0; j < 4; ++j) {  // Q tile 16x256
    int idx = tid + 256 * j;
    int r = idx >> 6, c4 = (idx & 63) * 4;
    const float4 q = *(const float4*)(Qp + (size_t)(b * SEQ + q0 + r) * DMODEL + c4);
    Qs[r][c4 + 0] = q.x; Qs[r][c4 + 1] = q.y;
    Qs[r][c4 + 2] = q.z; Qs[r][c4 + 3] = q.w;
  }
  if (tid < 64) {
    int hh = tid >> 4, r = tid & 15;
    size_t o = (size_t)(b * NHEAD + hh) * SEQ + q0 + r;
    lm[hh][r] = wm[o];
    lz[hh][r] = wz[o];
  }
  __syncthreads();

  float accv[8];
#pragma unroll
  for (int v = 0; v < 8; ++v) accv[v] = 0.f;

  for (int h = 0; h < NHEAD; ++h) {
#pragma unroll
    for (int j = 0; j < 8; ++j) {  // K head-slice 16x64 (wave-private)
      int idx = lane + 32 * j;
      int r = idx >> 4, c4 = (idx & 15) * 4;
      const float4 k4 = *(const float4*)(Kp + (size_t)(b * SEQ + key0 + r) * DMODEL + h * DK + c4);
      Ks[wv][r][c4 + 0] = k4.x; Ks[wv][r][c4 + 1] = k4.y;
      Ks[wv][r][c4 + 2] = k4.z; Ks[wv][r][c4 + 3] = k4.w;
    }
    __syncthreads();

    v8f s = v8f_zero();
#pragma unroll
    for (int ks = 0; ks < DK; ks += 4) {
      v2f a = ldPair(&Qs[0][0], 260, mr, h * DK + ks, hl);
      v2f bt = ldPair(&Ks[wv][0][0], 68, mr, ks, hl);
      s = wmma_f32(a, bt, s);
    }
#pragma unroll
    for (int v = 0; v < 8; ++v) {
      int rq = v + 8 * hl;
      float p = __expf(s[v] * INV_SCALE - lm[h][rq]) / lz[h][rq];
      accv[v] += 0.25f * p;
    }
    __syncthreads();
  }

  const int keyg = key0 + mr;
#pragma unroll
  for (int v = 0; v < 8; ++v) {
    avg[(size_t)(b * SEQ + q0 + v + 8 * hl) * SEQ + keyg] = accv[v];
  }
}

// ---------------------------------------------------------------------------
// ctx[b, 32 q-rows, 256] = avg_attn[b, q, 0:2048] @ V[b, 0:2048, 256].
// grid = (SEQ/32, BATCH), block = 256; wave w: rows (w&1)*16, cols (w>>1)*64.
// V tile staged TRANSPOSED in LDS: Vt[n][k].
// ---------------------------------------------------------------------------
__global__ __launch_bounds__(256) void ctx_gemm(const float* __restrict__ avg,
                                                const float* __restrict__ Vp,
                                                float* __restrict__ ctx) {
  __shared__ float As[32][18];     // [m][k]
  __shared__ float Vt[256][18];    // [n][k] transposed

  const int tid = threadIdx.x;
  const int lane = tid & 31;
  const int wv = tid >> 5;
  const int mr = lane & 15;
  const int hl = lane >> 4;
  const int q0 = blockIdx.x * 32;
  const int b = blockIdx.y;
  const int m0 = (wv & 1) * 16;
  const int nb = (wv >> 1) * 64;

  v8f acc[4];
#pragma unroll
  for (int j = 0; j < 4; ++j) acc[j] = v8f_zero();

  for (int k0 = 0; k0 < SEQ; k0 += 16) {
    {  // avg tile 32x16
      int r = tid >> 3, c2 = (tid & 7) * 2;
      const float2 a2 = *(const float2*)(avg + (size_t)(b * SEQ + q0 + r) * SEQ + k0 + c2);
      As[r][c2 + 0] = a2.x; As[r][c2 + 1] = a2.y;
      if (k0 + 16 < SEQ)
        __builtin_prefetch(avg + (size_t)(b * SEQ + q0 + r) * SEQ + k0 + 16 + c2, 0, 3);
    }
#pragma unroll
    for (int j = 0; j < 4; ++j) {  // V tile 16x256 -> transposed store
      int idx = tid + 256 * j;
      int r = idx >> 6, c4 = (idx & 63) * 4;
      const float4 v4 = *(const float4*)(Vp + (size_t)(b * SEQ + k0 + r) * DMODEL + c4);
      Vt[c4 + 0][r] = v4.x; Vt[c4 + 1][r] = v4.y;
      Vt[c4 + 2][r] = v4.z; Vt[c4 + 3][r] = v4.w;
      if (k0 + 16 < SEQ)
        __builtin_prefetch(Vp + (size_t)(b * SEQ + k0 + 16 + r) * DMODEL + c4, 0, 3);
    }
    __syncthreads();
#pragma unroll
    for (int ks = 0; ks < 16; ks += 4) {
      v2f a = ldPair(&As[0][0], 18, m0 + mr, ks, hl);
#pragma unroll
      for (int j = 0; j < 4; ++j) {
        v2f bfr = ldPair(&Vt[0][0], 18, nb + j * 16 + mr, ks, hl);
        acc[j] = wmma_f32(a, bfr, acc[j]);
      }
    }
    __syncthreads();
  }

#pragma unroll
  for (int j = 0; j < 4; ++j) {
#pragma unroll
    for (int v = 0; v < 8; ++v) {
      ctx[(size_t)(b * SEQ + q0 + m0 + v + 8 * hl) * DMODEL + nb + j * 16 + mr] = acc[j][v];
    }
  }
}

// ---------------------------------------------------------------------------
extern "C" void kernel_launch(void* const* d_in, const int* in_sizes, int n_in,
                              void* d_out, int out_size, void* d_ws, size_t ws_size,
                              hipStream_t stream) {
  (void)in_sizes; (void)n_in; (void)out_size; (void)ws_size;

  const float* query = (const float*)d_in[0];
  const float* key   = (const float*)d_in[1];
  const float* value = (const float*)d_in[2];
  const float* W_q = (const float*)d_in[3];
  const float* b_q = (const float*)d_in[4];
  const float* W_k = (const float*)d_in[5];
  const float* b_k = (const float*)d_in[6];
  const float* W_v = (const float*)d_in[7];
  const float* b_v = (const float*)d_in[8];
  const float* W_o = (const float*)d_in[9];
  const float* b_o = (const float*)d_in[10];

  const size_t BSD = (size_t)BATCH * SEQ * DMODEL;  // 4,194,304
  float* out = (float*)d_out;
  float* avg = out + BSD;  // avg_attn region of d_out (B,S,S)

  float* Qp  = (float*)d_ws;
  float* Kp  = Qp + BSD;
  float* Vp  = Kp + BSD;
  float* ctx = Vp + BSD;
  float* wsm = ctx + BSD;
  float* wsz = wsm + (size_t)BATCH * NHEAD * SEQ;

  const dim3 blk(256);
  const int Mrows = BATCH * SEQ;  // 16384

  // Projections: Q, K, V
  gemm_xw<<<dim3(Mrows / 64), blk, 0, stream>>>(query, W_q, b_q, Qp);
  gemm_xw<<<dim3(Mrows / 64), blk, 0, stream>>>(key,   W_k, b_k, Kp);
  gemm_xw<<<dim3(Mrows / 64), blk, 0, stream>>>(value, W_v, b_v, Vp);

  // Softmax row stats (m, Z) per (b, h, q)
  attn_stats<<<dim3(SEQ / 16, NHEAD, BATCH), blk, 0, stream>>>(Qp, Kp, wsm, wsz);

  // Head-averaged attention matrix -> d_out avg region
  attn_avg<<<dim3(SEQ / 128, SEQ / 16, BATCH), blk, 0, stream>>>(Qp, Kp, wsm, wsz, avg);

  // ctx = avg_attn @ V
  ctx_gemm<<<dim3(SEQ / 32, BATCH), blk, 0, stream>>>(avg, Vp, ctx);

  // out = ctx @ W_o + b_o
  gemm_xw<<<dim3(Mrows / 64), blk, 0, stream>>>(ctx, W_o, b_o, out);
}